// GAT_33715493273868
// MI455X (gfx1250) — compile-verified
//
#include <hip/hip_runtime.h>
#include <math.h>

// Problem constants (match reference)
#define NNODES 50000
#define NEDGES 800000
#define NH     4
#define NCCH   32
#define HIDDIM 128
#define LATDIM 32
#define NGRAPH 128
#define INDIM  64
#define ETOT   (NEDGES + NNODES)            // self-loops appended: 850000
#define NEGS   0.2f
#define CHUNK  256
#define NCH    ((NNODES + CHUNK - 1) / CHUNK)   // 196
#define PSTRIP 25

typedef __attribute__((ext_vector_type(2))) float v2f;
typedef __attribute__((ext_vector_type(8))) float v8f;

__device__ __forceinline__ float lrelu(float v) { return (v > 0.f) ? v : NEGS * v; }
__device__ __forceinline__ float sel4(const float4& v, int h) {
  float r = v.x;
  r = (h == 1) ? v.y : r;
  r = (h == 2) ? v.z : r;
  r = (h == 3) ? v.w : r;
  return r;
}

// ---------------------------------------------------------------------------
// WMMA fp32 GEMM: C[M, NT*16] = A[M,K] @ B[K, NT*16], row-major.
// Whole B panel staged into LDS once per block (<= 64KB, WGP has 320KB).
// One wave computes a 16-row strip across all NT column tiles (register tiled),
// issuing NT WMMAs per k-step. Dead waves stay alive for __syncthreads and
// keep EXEC all-ones through the WMMAs (A rows clamped, stores guarded).
// ---------------------------------------------------------------------------
template <int NT>
__global__ __launch_bounds__(256) void k_gemm_wmma(
    const float* __restrict__ A, const float* __restrict__ B,
    float* __restrict__ Cm, int M, int K)
{
  constexpr int Nc = NT * 16;
  __shared__ float Bs[128 * Nc];

  const int tid = threadIdx.x;
  const int nv4 = (K * Nc) >> 2;
  for (int i = tid; i < nv4; i += 256)
    ((float4*)Bs)[i] = ((const float4*)B)[i];
  __syncthreads();

  const int wave    = tid >> 5;
  const int lane    = tid & 31;
  const int half    = lane >> 4;     // selects K-pair per the 16x4 f32 A layout
  const int l16     = lane & 15;
  const int rowBase = (blockIdx.x * 8 + wave) * 16;
  const bool active = rowBase < M;
  const int rowT    = active ? rowBase : (M - 16);

  const float* aptr = A + (size_t)(rowT + l16) * K + 2 * half;

  v8f acc[NT] = {};
  for (int k0 = 0; k0 < K; k0 += 4) {
    v2f a = *(const v2f*)(aptr + k0);                 // A[row][k0+2h .. +1]
    const float* b0 = Bs + (k0 + 2 * half) * Nc + l16;
#pragma unroll
    for (int t = 0; t < NT; ++t) {
      v2f b;
      b.x = b0[t * 16];                               // B[k0+2h  ][t*16+col]
      b.y = b0[t * 16 + Nc];                          // B[k0+2h+1][t*16+col]
      acc[t] = __builtin_amdgcn_wmma_f32_16x16x4_f32(
          false, a, false, b, (short)0, acc[t], false, false);
    }
  }

  if (active) {
    float* cp0 = Cm + (size_t)(rowBase + half * 8) * Nc + l16;
#pragma unroll
    for (int t = 0; t < NT; ++t)
#pragma unroll
      for (int v = 0; v < 8; ++v)
        cp0[(size_t)v * Nc + t * 16] = acc[t][v];
  }
}

// ---------------------------------------------------------------------------
// alpha_s / alpha_d per (node, head)
// ---------------------------------------------------------------------------
__global__ __launch_bounds__(256) void k_attn(
    const float* __restrict__ Hf, const float* __restrict__ aS,
    const float* __restrict__ aD, float* __restrict__ alS, float* __restrict__ alD)
{
  int idx = blockIdx.x * 256 + threadIdx.x;
  if (idx >= NNODES * NH) return;
  int n = idx >> 2, h = idx & 3;
  const float* hp = Hf + (size_t)n * HIDDIM + h * NCCH;
  float s = 0.f, d = 0.f;
#pragma unroll
  for (int c = 0; c < NCCH; ++c) {
    float v = hp[c];
    s = fmaf(v, aS[h * NCCH + c], s);
    d = fmaf(v, aD[h * NCCH + c], d);
  }
  alS[idx] = s;
  alD[idx] = d;
}

__global__ __launch_bounds__(256) void k_zero(float* p, int n)
{
  int i = blockIdx.x * 256 + threadIdx.x;
  if (i < n) p[i] = 0.f;
}

// --------------------------- CSR build (by dst) ----------------------------
__global__ __launch_bounds__(256) void k_count(const int* __restrict__ dstA,
                                               int* __restrict__ deg)
{
  int ei = blockIdx.x * 256 + threadIdx.x;
  if (ei >= ETOT) return;
  int d = (ei < NEDGES) ? dstA[ei] : (ei - NEDGES);
  atomicAdd(&deg[d], 1);
}

__global__ __launch_bounds__(256) void k_chunksum(const int* __restrict__ deg,
                                                  int* __restrict__ csum)
{
  int c = blockIdx.x * 256 + threadIdx.x;
  if (c >= NCH) return;
  int n0 = c * CHUNK, n1 = min(n0 + CHUNK, NNODES);
  int s = 0;
  for (int i = n0; i < n1; ++i) s += deg[i];
  csum[c] = s;
}

__global__ void k_chunkscan(int* csum)
{
  if (threadIdx.x != 0 || blockIdx.x != 0) return;
  int run = 0;
  for (int c = 0; c < NCH; ++c) { int v = csum[c]; csum[c] = run; run += v; }
}

__global__ __launch_bounds__(256) void k_scanwrite(
    const int* __restrict__ deg, const int* __restrict__ csum,
    int* __restrict__ rowPtr, int* __restrict__ cursor)
{
  int c = blockIdx.x * 256 + threadIdx.x;
  if (c >= NCH) return;
  int n0 = c * CHUNK, n1 = min(n0 + CHUNK, NNODES);
  int base = csum[c];
  for (int i = n0; i < n1; ++i) { rowPtr[i] = base; cursor[i] = base; base += deg[i]; }
  if (c == NCH - 1) rowPtr[NNODES] = base;
}

__global__ __launch_bounds__(256) void k_fill(const int* __restrict__ dstA,
                                              int* __restrict__ cursor,
                                              int* __restrict__ eAdj)
{
  int ei = blockIdx.x * 256 + threadIdx.x;
  if (ei >= ETOT) return;
  int d = (ei < NEDGES) ? dstA[ei] : (ei - NEDGES);
  int pos = atomicAdd(&cursor[d], 1);
  eAdj[pos] = ei;
}

// ---------------------------------------------------------------------------
// Fused per-dst-node GAT aggregation. One wave32 per node:
//   pass 1: lane-parallel per-head max of leakyrelu(alS[src]+alD[n]) + shfl reduce
//   pass 2: lane-parallel per-head sum(exp(e - m))                   + shfl reduce
//   pass 3: edge-sequential; 32 lanes x float4 = 128 channels; acc in registers
//   epilogue: + bias, leaky-relu, write activation. Zero atomics.
// ---------------------------------------------------------------------------
__global__ __launch_bounds__(256) void k_aggregate(
    const int* __restrict__ srcA, const int* __restrict__ rowPtr,
    const int* __restrict__ eAdj, const float* __restrict__ alS,
    const float* __restrict__ alD, const float* __restrict__ Hf,
    const float* __restrict__ bias, float* __restrict__ act)
{
  const int n = blockIdx.x * 8 + (threadIdx.x >> 5);
  if (n >= NNODES) return;                 // wave-uniform
  const int lane = threadIdx.x & 31;
  const int beg = rowPtr[n], end = rowPtr[n + 1];

  const float4 adn = *(const float4*)(alD + (size_t)n * NH);

  float4 mx = make_float4(-INFINITY, -INFINITY, -INFINITY, -INFINITY);
  for (int j = beg + lane; j < end; j += 32) {
    int ei = eAdj[j];
    int s  = (ei < NEDGES) ? srcA[ei] : (ei - NEDGES);
    float4 a4 = *(const float4*)(alS + (size_t)s * NH);
    mx.x = fmaxf(mx.x, lrelu(a4.x + adn.x));
    mx.y = fmaxf(mx.y, lrelu(a4.y + adn.y));
    mx.z = fmaxf(mx.z, lrelu(a4.z + adn.z));
    mx.w = fmaxf(mx.w, lrelu(a4.w + adn.w));
  }
#pragma unroll
  for (int o = 16; o; o >>= 1) {
    mx.x = fmaxf(mx.x, __shfl_xor(mx.x, o));
    mx.y = fmaxf(mx.y, __shfl_xor(mx.y, o));
    mx.z = fmaxf(mx.z, __shfl_xor(mx.z, o));
    mx.w = fmaxf(mx.w, __shfl_xor(mx.w, o));
  }

  float4 sm = make_float4(0.f, 0.f, 0.f, 0.f);
  for (int j = beg + lane; j < end; j += 32) {
    int ei = eAdj[j];
    int s  = (ei < NEDGES) ? srcA[ei] : (ei - NEDGES);
    float4 a4 = *(const float4*)(alS + (size_t)s * NH);
    sm.x += expf(lrelu(a4.x + adn.x) - mx.x);
    sm.y += expf(lrelu(a4.y + adn.y) - mx.y);
    sm.z += expf(lrelu(a4.z + adn.z) - mx.z);
    sm.w += expf(lrelu(a4.w + adn.w) - mx.w);
  }
#pragma unroll
  for (int o = 16; o; o >>= 1) {
    sm.x += __shfl_xor(sm.x, o);
    sm.y += __shfl_xor(sm.y, o);
    sm.z += __shfl_xor(sm.z, o);
    sm.w += __shfl_xor(sm.w, o);
  }

  const int h     = lane >> 3;                  // 8 lanes per 32-channel head
  const float mh  = sel4(mx, h);
  const float rdh = 1.f / (sel4(sm, h) + 1e-16f);
  const float adh = sel4(adn, h);

  float4 accv = make_float4(0.f, 0.f, 0.f, 0.f);
  for (int j = beg; j < end; ++j) {
    int ei = eAdj[j];
    int s  = (ei < NEDGES) ? srcA[ei] : (ei - NEDGES);
    float a = expf(lrelu(alS[s * NH + h] + adh) - mh) * rdh;
    float4 hv = *(const float4*)(Hf + (size_t)s * HIDDIM + lane * 4);
    accv.x = fmaf(a, hv.x, accv.x);
    accv.y = fmaf(a, hv.y, accv.y);
    accv.z = fmaf(a, hv.z, accv.z);
    accv.w = fmaf(a, hv.w, accv.w);
  }

  const float4 b4 = *(const float4*)(bias + lane * 4);
  accv.x = lrelu(accv.x + b4.x);
  accv.y = lrelu(accv.y + b4.y);
  accv.z = lrelu(accv.z + b4.z);
  accv.w = lrelu(accv.w + b4.w);
  *(float4*)(act + (size_t)n * HIDDIM + lane * 4) = accv;
}

// --------------------- pooling (batch sorted -> strips) ---------------------
__global__ __launch_bounds__(128) void k_pool(
    const float* __restrict__ act, const int* __restrict__ batch,
    float* __restrict__ pooled)
{
  int n0 = blockIdx.x * PSTRIP;
  if (n0 >= NNODES) return;
  int n1 = min(n0 + PSTRIP, NNODES);
  int ch = threadIdx.x;
  int g = batch[n0];
  float s = 0.f;
  for (int n = n0; n < n1; ++n) {
    int gn = batch[n];
    if (gn != g) { atomicAdd(&pooled[(size_t)g * HIDDIM + ch], s); s = 0.f; g = gn; }
    s += act[(size_t)n * HIDDIM + ch];
  }
  atomicAdd(&pooled[(size_t)g * HIDDIM + ch], s);
}

// out = bn_g * (tmp + fc_b) / sqrt(1+eps) + bn_b
__global__ __launch_bounds__(256) void k_head(
    const float* __restrict__ tmp, const float* __restrict__ fcb,
    const float* __restrict__ g, const float* __restrict__ bb,
    float* __restrict__ out)
{
  int idx = blockIdx.x * 256 + threadIdx.x;
  if (idx >= NGRAPH * LATDIM) return;
  int c = idx & (LATDIM - 1);
  float v = (tmp[idx] + fcb[c]) * 0.9999950000374997f;   // 1/sqrt(1+1e-5)
  out[idx] = g[c] * v + bb[c];
}

// ---------------------------------------------------------------------------
extern "C" void kernel_launch(void* const* d_in, const int* in_sizes, int n_in,
                              void* d_out, int out_size, void* d_ws, size_t ws_size,
                              hipStream_t stream)
{
  (void)in_sizes; (void)n_in; (void)out_size; (void)ws_size;
  const float* x     = (const float*)d_in[0];
  const int*   eidx  = (const int*)d_in[1];
  const int*   batch = (const int*)d_in[2];
  const float* Wm[3] = {(const float*)d_in[3], (const float*)d_in[7],  (const float*)d_in[11]};
  const float* As[3] = {(const float*)d_in[4], (const float*)d_in[8],  (const float*)d_in[12]};
  const float* Ad[3] = {(const float*)d_in[5], (const float*)d_in[9],  (const float*)d_in[13]};
  const float* Bi[3] = {(const float*)d_in[6], (const float*)d_in[10], (const float*)d_in[14]};
  const float* fcw = (const float*)d_in[15];
  const float* fcb = (const float*)d_in[16];
  const float* bng = (const float*)d_in[17];
  const float* bnb = (const float*)d_in[18];

  // workspace layout
  float* Hf     = (float*)d_ws;                          // N*HID
  float* act    = Hf  + (size_t)NNODES * HIDDIM;         // N*HID
  float* alS    = act + (size_t)NNODES * HIDDIM;         // N*H
  float* alD    = alS + (size_t)NNODES * NH;             // N*H
  int*   deg    = (int*)(alD + (size_t)NNODES * NH);     // N
  int*   rowPtr = deg + NNODES;                          // N+4 (padded)
  int*   cursor = rowPtr + (NNODES + 4);                 // N
  int*   eAdj   = cursor + NNODES;                       // ETOT
  int*   csum   = eAdj + ETOT;                           // 256 (padded)
  float* pooled = (float*)(csum + 256);                  // G*HID
  float* htmp   = pooled + (size_t)NGRAPH * HIDDIM;      // G*LAT

  const int* srcA = eidx;
  const int* dstA = eidx + NEDGES;

  const int mtiles = NNODES / 16;                        // 3125 exact
  const int gemmBx = (mtiles + 7) / 8;                   // 391
  const int nhTot  = NNODES * NH;

  // ---- CSR by dst (rebuilt every call; graph inputs are never mutated) ----
  k_zero<<<(NNODES + 255) / 256, 256, 0, stream>>>((float*)deg, NNODES);
  k_count<<<(ETOT + 255) / 256, 256, 0, stream>>>(dstA, deg);
  k_chunksum<<<(NCH + 255) / 256, 256, 0, stream>>>(deg, csum);
  k_chunkscan<<<1, 32, 0, stream>>>(csum);
  k_scanwrite<<<(NCH + 255) / 256, 256, 0, stream>>>(deg, csum, rowPtr, cursor);
  k_fill<<<(ETOT + 255) / 256, 256, 0, stream>>>(dstA, cursor, eAdj);

  // ---- 3 GAT layers ----
  for (int L = 0; L < 3; ++L) {
    const float* inp = (L == 0) ? x : act;
    const int K      = (L == 0) ? INDIM : HIDDIM;

    k_gemm_wmma<8><<<gemmBx, 256, 0, stream>>>(inp, Wm[L], Hf, NNODES, K);
    k_attn<<<(nhTot + 255) / 256, 256, 0, stream>>>(Hf, As[L], Ad[L], alS, alD);
    k_aggregate<<<(NNODES + 7) / 8, 256, 0, stream>>>(
        srcA, rowPtr, eAdj, alS, alD, Hf, Bi[L], act);
  }

  // ---- pool + FC head (WMMA) + BN ----
  k_zero<<<(NGRAPH * HIDDIM + 255) / 256, 256, 0, stream>>>(pooled, NGRAPH * HIDDIM);
  k_pool<<<(NNODES + PSTRIP - 1) / PSTRIP, 128, 0, stream>>>(act, batch, pooled);
  k_gemm_wmma<2><<<1, 256, 0, stream>>>(pooled, fcw, htmp, NGRAPH, HIDDIM);
  k_head<<<(NGRAPH * LATDIM + 255) / 256, 256, 0, stream>>>(htmp, fcb, bng, bnb, (float*)d_out);
}